// MLPModelWPathways_89541478187023
// MI455X (gfx1250) — compile-verified
//
#include <hip/hip_runtime.h>
#include <hip/hip_bf16.h>

// ---------------------------------------------------------------------------
// MLPModelWPathways forward for MI455X (gfx1250, wave32, WMMA).
// f32 end to end: both big GEMMs are bandwidth-bound (~30 flop/byte vs the
// 23.3 TB/s HBM roof), so V_WMMA_F32_16X16X4_F32 is the right matrix op.
// ---------------------------------------------------------------------------

typedef float v2f __attribute__((ext_vector_type(2)));
typedef float v8f __attribute__((ext_vector_type(8)));

constexpr int BS        = 4;
constexpr int N_GENES   = 15135;
constexpr int N_FEAT    = 64;
constexpr int HIDDEN    = 128;
constexpr int HIDDEN_FC = 256;
constexpr int N_CMT     = 1000;
constexpr int N_CLASSES = 10;

constexpr int M_ROWS  = BS * N_GENES;        // 60540 fused (b, gene) rows
constexpr int FEAT_CH = 2 * HIDDEN;          // 256 interleaved channels
constexpr int M_TILES = (M_ROWS + 15) / 16;  // 3784

// ---------------------------------------------------------------------------
// Unified WMMA GEMM + bias + relu, striped store into feat.
//   C[gm*FEAT_CH + 2*n + COFF] = relu(sum_k A[gm*lda + k*ASTEP] * B[k*128+n] + bias[n])
// Block: 256 threads = 8 waves; block owns a 16-row M tile, wave w owns the
// 16-col N tile w (N = 128 total). K consumed in 64-wide LDS panels,
// 16 v_wmma_f32_16x16x4_f32 steps per panel.
// ---------------------------------------------------------------------------
template<int K, int ASTEP, int COFF>
__global__ __launch_bounds__(256) void gemm_relu_wmma(
    const float* __restrict__ A, int lda,
    const float* __restrict__ B,      // [K, 128] row-major (din, dout)
    const float* __restrict__ bias,   // [128]
    float* __restrict__ C)            // feat [M_ROWS, 256]
{
    __shared__ float BsT[HIDDEN][64 + 1];  // B panel transposed: BsT[n][k]
    __shared__ float As[16][64];           // A panel: As[m][k]

    const int tid  = threadIdx.x;
    const int lane = tid & 31;
    const int wave = tid >> 5;             // 0..7 == N tile
    const int m0   = blockIdx.x * 16;

    v8f acc = {};

    for (int kb = 0; kb < K; kb += 64) {
        // Stage B panel [64 x 128] transposed (coalesced global reads).
        for (int i = tid; i < 64 * HIDDEN; i += 256) {
            const int k = i >> 7;          // 0..63
            const int n = i & 127;
            BsT[n][k] = B[(kb + k) * HIDDEN + n];
        }
        // Stage A panel [16 x 64]; zero-pad the ragged last M tile.
        for (int i = tid; i < 16 * 64; i += 256) {
            const int m  = i >> 6;
            const int k  = i & 63;
            const int gm = m0 + m;
            As[m][k] = (gm < M_ROWS)
                     ? A[(size_t)gm * lda + (size_t)(kb + k) * ASTEP]
                     : 0.0f;
        }
        __syncthreads();

        // ISA A 16x4 f32 layout: lane&15 = M; lanes<16 hold K+{0,1},
        // lanes>=16 hold K+{2,3}. B 4x16 is symmetric with N = lane&15.
        const int base = (lane >> 4) << 1;          // 0 or 2
        const int ml   = lane & 15;
        const int nl   = wave * 16 + (lane & 15);
        #pragma unroll
        for (int kk = 0; kk < 16; ++kk) {
            const int k0 = kk * 4 + base;
            v2f a, b;
            a.x = As[ml][k0];
            a.y = As[ml][k0 + 1];
            b.x = BsT[nl][k0];
            b.y = BsT[nl][k0 + 1];
            acc = __builtin_amdgcn_wmma_f32_16x16x4_f32(
                /*neg_a=*/false, a, /*neg_b=*/false, b,
                /*c_mod=*/(short)0, acc, /*reuse_a=*/false, /*reuse_b=*/false);
        }
        __syncthreads();
    }

    // C/D layout: VGPR v holds M=v (lanes 0-15) / M=v+8 (lanes 16-31), N=lane&15.
    const int   nl    = wave * 16 + (lane & 15);
    const float bn    = bias[nl];
    const int   mbase = (lane >> 4) << 3;           // 0 or 8
    #pragma unroll
    for (int v = 0; v < 8; ++v) {
        const int gm = m0 + mbase + v;
        if (gm < M_ROWS) {
            float r = acc[v] + bn;
            r = r > 0.0f ? r : 0.0f;
            C[(size_t)gm * FEAT_CH + 2 * nl + COFF] = r;
        }
    }
}

// ---------------------------------------------------------------------------
// Segment-sum helpers (feat + seg live in L2: 62 MB + 4 MB << 192 MB).
// ---------------------------------------------------------------------------
__global__ void zero_kernel(float* __restrict__ p, size_t n) {
    const size_t i = (size_t)blockIdx.x * blockDim.x + threadIdx.x;
    if (i < n) p[i] = 0.0f;
}

__global__ void count_edges(const int* __restrict__ col,
                            float* __restrict__ cnt, int n_edges) {
    const int e = blockIdx.x * blockDim.x + threadIdx.x;
    if (e < n_edges) atomicAdd(&cnt[col[e]], 1.0f);
}

// One block per edge: scatter feat[:, row[e], :] (bs*256 = 1024 floats)
// into seg[col[e]][b][ch].
__global__ __launch_bounds__(256) void scatter_edges(
    const float* __restrict__ feat, const int* __restrict__ row,
    const int* __restrict__ col, float* __restrict__ seg, int n_edges)
{
    const int e = blockIdx.x;
    if (e >= n_edges) return;
    const int r = row[e];
    const int c = col[e];
    const int tid = threadIdx.x;
    #pragma unroll
    for (int i = 0; i < 4; ++i) {
        const int idx = tid + i * 256;          // idx = b*256 + ch
        const int b   = idx >> 8;
        const int ch  = idx & 255;
        const float v = feat[((size_t)b * N_GENES + r) * FEAT_CH + ch];
        atomicAdd(&seg[(size_t)c * (BS * FEAT_CH) + idx], v);
    }
}

// scores[b,c] = dot(seg[c][b][:], Wfc) / max(cnt[c],1) + bfc
__global__ void scores_kernel(const float* __restrict__ seg,
                              const float* __restrict__ cnt,
                              const float* __restrict__ Wfc,
                              const float* __restrict__ bfc,
                              float* __restrict__ scores) {
    const int i = blockIdx.x * blockDim.x + threadIdx.x;
    if (i >= BS * N_CMT) return;
    const int b = i / N_CMT;
    const int c = i % N_CMT;
    const float* s = seg + (size_t)c * (BS * FEAT_CH) + b * FEAT_CH;
    float acc = 0.0f;
    for (int ch = 0; ch < FEAT_CH; ++ch) acc += s[ch] * Wfc[ch];
    float d = cnt[c];
    d = d > 1.0f ? d : 1.0f;
    scores[b * N_CMT + c] = acc / d + bfc[0];
}

// Head: relu(scores @ Wl1 + bl1) @ Wl2 + bl2, then row log_softmax. 4 rows.
__global__ __launch_bounds__(256) void head_kernel(
    const float* __restrict__ scores, const float* __restrict__ Wl1,
    const float* __restrict__ bl1, const float* __restrict__ Wl2,
    const float* __restrict__ bl2, float* __restrict__ out)
{
    __shared__ float h[BS][HIDDEN_FC];
    __shared__ float logit[BS][N_CLASSES];
    const int tid = threadIdx.x;        // 256 threads, tid == hidden unit j
    for (int b = 0; b < BS; ++b) {
        float acc = bl1[tid];
        for (int c = 0; c < N_CMT; ++c)
            acc += scores[b * N_CMT + c] * Wl1[c * HIDDEN_FC + tid];
        h[b][tid] = acc > 0.0f ? acc : 0.0f;
    }
    __syncthreads();
    if (tid < BS * N_CLASSES) {
        const int b = tid / N_CLASSES;
        const int k = tid % N_CLASSES;
        float acc = bl2[k];
        for (int j = 0; j < HIDDEN_FC; ++j)
            acc += h[b][j] * Wl2[j * N_CLASSES + k];
        logit[b][k] = acc;
    }
    __syncthreads();
    if (tid < BS) {
        float mx = logit[tid][0];
        for (int k = 1; k < N_CLASSES; ++k) mx = fmaxf(mx, logit[tid][k]);
        float s = 0.0f;
        for (int k = 0; k < N_CLASSES; ++k) s += __expf(logit[tid][k] - mx);
        const float lse = mx + __logf(s);
        for (int k = 0; k < N_CLASSES; ++k)
            out[tid * N_CLASSES + k] = logit[tid][k] - lse;
    }
}

// ---------------------------------------------------------------------------
extern "C" void kernel_launch(void* const* d_in, const int* in_sizes, int n_in,
                              void* d_out, int out_size, void* d_ws, size_t ws_size,
                              hipStream_t stream) {
    (void)n_in; (void)out_size; (void)ws_size;

    const float* x   = (const float*)d_in[0];
    // d_in[1] = batch (unused by the forward pass)
    const int*   row = (const int*)  d_in[2];
    const int*   col = (const int*)  d_in[3];
    const float* W1  = (const float*)d_in[4];
    const float* b1  = (const float*)d_in[5];
    const float* W2  = (const float*)d_in[6];
    const float* b2  = (const float*)d_in[7];
    const float* Wfc = (const float*)d_in[8];
    const float* bfc = (const float*)d_in[9];
    const float* Wl1 = (const float*)d_in[10];
    const float* bl1 = (const float*)d_in[11];
    const float* Wl2 = (const float*)d_in[12];
    const float* bl2 = (const float*)d_in[13];
    const int n_edges = in_sizes[2];

    // Workspace layout (floats): feat | seg | cnt | scores  (~66 MB total)
    float* feat   = (float*)d_ws;                          // M_ROWS * 256
    float* seg    = feat + (size_t)M_ROWS * FEAT_CH;       // N_CMT * BS * 256
    float* cnt    = seg  + (size_t)N_CMT * BS * FEAT_CH;   // N_CMT
    float* scores = cnt  + N_CMT;                          // BS * N_CMT

    // Fresh zeros every call (graph-replay safe; atomics accumulate below).
    const size_t nz = (size_t)N_CMT * BS * FEAT_CH + N_CMT;
    zero_kernel<<<(int)((nz + 255) / 256), 256, 0, stream>>>(seg, nz);

    // h1 = relu(x @ W1 + b1)  -> feat even channels
    gemm_relu_wmma<N_FEAT, 1, 0><<<M_TILES, 256, 0, stream>>>(
        x, N_FEAT, W1, b1, feat);
    // h2 = relu(h1 @ W2 + b2) -> feat odd channels (reads even channels)
    gemm_relu_wmma<HIDDEN, 2, 1><<<M_TILES, 256, 0, stream>>>(
        feat, FEAT_CH, W2, b2, feat);

    // Segment counts + sums over edges.
    count_edges<<<(n_edges + 255) / 256, 256, 0, stream>>>(col, cnt, n_edges);
    scatter_edges<<<n_edges, 256, 0, stream>>>(feat, row, col, seg, n_edges);

    // Pathway scores and FC head.
    scores_kernel<<<(BS * N_CMT + 255) / 256, 256, 0, stream>>>(
        seg, cnt, Wfc, bfc, scores);
    head_kernel<<<1, 256, 0, stream>>>(scores, Wl1, bl1, Wl2, bl2,
                                       (float*)d_out);
}